// PostProcessor_24653112279514
// MI455X (gfx1250) — compile-verified
//
#include <hip/hip_runtime.h>
#include <math.h>

#define BB    2
#define PP    512
#define CC    21
#define CFG   (CC - 1)        // 20 foreground classes
#define NSETS (BB * CFG)      // 40
#define DETN  100
#define IMW_F 1024.0f
#define IMH_F 800.0f
#define SCORE_THRESH 0.05f
#define NMS_THR 0.5f
#define PI_F 3.14159265358979323846f

// ---------------------------------------------------------------------------
// TDM (Tensor Data Mover) helpers — CDNA5 async tensor load to LDS
// ---------------------------------------------------------------------------
typedef __attribute__((ext_vector_type(4))) unsigned v4u;
typedef __attribute__((ext_vector_type(8))) int      v8i;
typedef __attribute__((ext_vector_type(4))) int      v4i;

#if __has_builtin(__builtin_amdgcn_tensor_load_to_lds)
#define HAVE_TDM 1
#else
#define HAVE_TDM 0
#endif

__device__ __forceinline__ void wait_tensor0() {
#if __has_builtin(__builtin_amdgcn_s_wait_tensorcnt)
  __builtin_amdgcn_s_wait_tensorcnt(0);
#else
  asm volatile("s_wait_tensorcnt 0" ::: "memory");
#endif
}

#if HAVE_TDM
// 1-D contiguous DWORD copy: global -> LDS via TDM descriptor (ISA 8.3/8.4).
// ndwords must fit in 16 bits (tile_dim0).
__device__ __forceinline__ void tdm_load_1d(void* lds_dst, const void* gsrc,
                                            unsigned ndwords) {
  unsigned lds = (unsigned)(uintptr_t)lds_dst;         // low 32 bits = LDS offset
  unsigned long long ga = (unsigned long long)(uintptr_t)gsrc;

  v4u g0;
  g0.x = 1u;                                           // count=1, user descriptor
  g0.y = lds;                                          // lds_addr
  g0.z = (unsigned)(ga & 0xFFFFFFFFull);               // global_addr[31:0]
  g0.w = (unsigned)((ga >> 32) & 0x01FFFFFFull)        // global_addr[56:32]
         | (2u << 30);                                 // type=2 ("image")

  v8i g1;
  g1[0] = (int)(2u << 16);                             // wg_mask=0, data_size=4B
  g1[1] = (int)((ndwords & 0xFFFFu) << 16);            // tensor_dim0[15:0]
  g1[2] = (int)((ndwords >> 16) | (1u << 16));         // tensor_dim0[31:16] | tensor_dim1=1
  g1[3] = (int)((ndwords & 0xFFFFu) << 16);            // tile_dim0 = ndwords
  g1[4] = (int)1;                                      // tile_dim1=1, tile_dim2=0
  g1[5] = (int)ndwords;                                // tensor_dim0_stride lo32
  g1[6] = 0;
  g1[7] = 0;

  v4i z4 = {0, 0, 0, 0};
#if defined(__clang_major__) && __clang_major__ >= 23
  v8i z8 = {0, 0, 0, 0, 0, 0, 0, 0};
  __builtin_amdgcn_tensor_load_to_lds(g0, g1, z4, z4, z8, 0);
#else
  __builtin_amdgcn_tensor_load_to_lds(g0, g1, z4, z4, 0);
#endif
}
#endif

// ---------------------------------------------------------------------------
// Geometry: exact convex quad-quad intersection area (Sutherland-Hodgman)
// p/q: interleaved x,y for 4 corners. Max vertices after 4 clips = 8.
// ---------------------------------------------------------------------------
__device__ float quad_inter_area(const float p[8], const float q[8]) {
  float poly[16], buf[16];
  int n = 4;
#pragma unroll
  for (int i = 0; i < 8; ++i) poly[i] = p[i];

  for (int k = 0; k < 4; ++k) {
    int k1 = (k + 1) & 3;
    float ax = q[2 * k], ay = q[2 * k + 1];
    float dx = q[2 * k1] - ax, dy = q[2 * k1 + 1] - ay;
    int m = 0;
    for (int i = 0; i < n; ++i) {
      int i1 = (i + 1 == n) ? 0 : i + 1;
      float sx = poly[2 * i], sy = poly[2 * i + 1];
      float ex = poly[2 * i1], ey = poly[2 * i1 + 1];
      float cs = dx * (sy - ay) - dy * (sx - ax);
      float ce = dx * (ey - ay) - dy * (ex - ax);
      bool s_in = (cs >= 0.0f), e_in = (ce >= 0.0f);
      if (s_in != e_in) {
        float den = cs - ce;
        den = (fabsf(den) > 1e-12f) ? den : 1e-12f;
        float t = cs / den;
        buf[2 * m] = sx + t * (ex - sx);
        buf[2 * m + 1] = sy + t * (ey - sy);
        ++m;
      }
      if (e_in) {
        buf[2 * m] = ex;
        buf[2 * m + 1] = ey;
        ++m;
      }
    }
    n = m;
    for (int i = 0; i < 2 * m; ++i) poly[i] = buf[i];
    if (n == 0) break;
  }
  if (n < 3) return 0.0f;
  float s = 0.0f;
  for (int i = 0; i < n; ++i) {
    int j = (i + 1 == n) ? 0 : i + 1;
    s += poly[2 * i] * poly[2 * j + 1] - poly[2 * j] * poly[2 * i + 1];
  }
  return 0.5f * fabsf(s);
}

// ---------------------------------------------------------------------------
// K1: one thread per proposal; softmax once, then loop the 20 fg classes:
// rotated decode + corner AABB + masked score, written set-major.
// ---------------------------------------------------------------------------
__global__ void decode_score_kernel(const float* __restrict__ logits,
                                    const float* __restrict__ reg,
                                    const float* __restrict__ anchors,
                                    float* __restrict__ scores_u,
                                    float* __restrict__ box5_u,
                                    float* __restrict__ bb4_u) {
  int n = blockIdx.x * blockDim.x + threadIdx.x;  // 0 .. BB*PP-1
  if (n >= BB * PP) return;

  // stable softmax denominator (once per proposal)
  const float* L = logits + n * CC;
  float lg[CC];
#pragma unroll
  for (int i = 0; i < CC; ++i) lg[i] = L[i];
  float mx = lg[0];
#pragma unroll
  for (int i = 1; i < CC; ++i) mx = fmaxf(mx, lg[i]);
  float den = 0.0f;
#pragma unroll
  for (int i = 0; i < CC; ++i) den += expf(lg[i] - mx);
  float rden = 1.0f / den;

  const float* a = anchors + (size_t)n * 5;
  float xc = a[0], yc = a[1], w = a[2], h = a[3], ang = a[4];
  const float clipv = logf(1000.0f / 16.0f);
  const float ox[4] = {-1.0f, 1.0f, 1.0f, -1.0f};
  const float oy[4] = {-1.0f, -1.0f, 1.0f, 1.0f};

  int b = n / PP, p = n - b * PP;

  for (int cf = 0; cf < CFG; ++cf) {
    int c = cf + 1;
    float sc = expf(lg[c] - mx) * rden;

    const float* d = reg + (size_t)n * CC * 5 + c * 5;
    float dx = d[0] * 0.1f, dy = d[1] * 0.1f;
    float dw = fminf(d[2] * 0.2f, clipv);
    float dh = fminf(d[3] * 0.2f, clipv);
    float da = d[4] * (1.0f / 3.0f);

    float px = dx * w + xc;
    float py = dy * h + yc;
    float pw = expf(dw) * w;
    float ph = expf(dh) * h;
    float pa = da * (180.0f / PI_F) + ang;

    float t = pa * (PI_F / 180.0f);
    float ct = cosf(t), st = sinf(t);
    float hx = pw * 0.5f, hy = ph * 0.5f;
    float mnx = 1e30f, mny = 1e30f, mxx = -1e30f, mxy = -1e30f;
#pragma unroll
    for (int i = 0; i < 4; ++i) {
      float X = px + ct * (ox[i] * hx) - st * (oy[i] * hy);
      float Y = py + st * (ox[i] * hx) + ct * (oy[i] * hy);
      mnx = fminf(mnx, X); mxx = fmaxf(mxx, X);
      mny = fminf(mny, Y); mxy = fmaxf(mxy, Y);
    }
    mnx = fminf(fmaxf(mnx, 0.0f), IMW_F - 1.0f);
    mxx = fminf(fmaxf(mxx, 0.0f), IMW_F - 1.0f);
    mny = fminf(fmaxf(mny, 0.0f), IMH_F - 1.0f);
    mxy = fminf(fmaxf(mxy, 0.0f), IMH_F - 1.0f);

    int set = b * CFG + cf;
    size_t sp = (size_t)set * PP + p;
    scores_u[sp] = (sc > SCORE_THRESH) ? sc : -1.0f;
    float* b5 = box5_u + sp * 5;
    b5[0] = px; b5[1] = py; b5[2] = pw; b5[3] = ph; b5[4] = pa;
    float* b4 = bb4_u + sp * 4;
    b4[0] = mnx; b4[1] = mny; b4[2] = mxx; b4[3] = mxy;
  }
}

// ---------------------------------------------------------------------------
// K2: per-set bitonic sort (score desc, index asc == stable argsort), gather,
//     precompute corner points + areas for the IoU stage.
// ---------------------------------------------------------------------------
__global__ void sort512_kernel(const float* __restrict__ scores_u,
                               const float* __restrict__ box5_u,
                               const float* __restrict__ bb4_u,
                               float* __restrict__ sc_s,
                               float* __restrict__ box5_s,
                               float* __restrict__ bb4_s,
                               float* __restrict__ pts_s,
                               float* __restrict__ area_s) {
  int set = blockIdx.x;
  int tid = threadIdx.x;  // 512
  __shared__ float kk[PP];
  __shared__ int   id[PP];
  kk[tid] = scores_u[(size_t)set * PP + tid];
  id[tid] = tid;
  __syncthreads();

  for (int k = 2; k <= PP; k <<= 1) {
    for (int j = k >> 1; j > 0; j >>= 1) {
      int i = tid, ixj = i ^ j;
      if (ixj > i) {
        float ka = kk[i], kb = kk[ixj];
        int ia = id[i], ib = id[ixj];
        bool before = (ka > kb) || (ka == kb && ia < ib);
        bool dir = ((i & k) == 0);
        if (before != dir) {
          kk[i] = kb; kk[ixj] = ka;
          id[i] = ib; id[ixj] = ia;
        }
      }
      __syncthreads();
    }
  }

  int src = id[tid];
  float sv = kk[tid];
  size_t dp = (size_t)set * PP + tid;
  size_t spn = (size_t)set * PP + src;
  sc_s[dp] = sv;
  float b5[5];
#pragma unroll
  for (int j = 0; j < 5; ++j) { b5[j] = box5_u[spn * 5 + j]; box5_s[dp * 5 + j] = b5[j]; }
#pragma unroll
  for (int j = 0; j < 4; ++j) bb4_s[dp * 4 + j] = bb4_u[spn * 4 + j];

  float t = b5[4] * (PI_F / 180.0f);
  float ct = cosf(t), st = sinf(t);
  float hx = b5[2] * 0.5f, hy = b5[3] * 0.5f;
  const float ox[4] = {-1.0f, 1.0f, 1.0f, -1.0f};
  const float oy[4] = {-1.0f, -1.0f, 1.0f, 1.0f};
  float* pt = pts_s + dp * 8;
#pragma unroll
  for (int i = 0; i < 4; ++i) {
    pt[2 * i]     = b5[0] + ct * (ox[i] * hx) - st * (oy[i] * hy);
    pt[2 * i + 1] = b5[1] + st * (ox[i] * hx) + ct * (oy[i] * hy);
  }
  area_s[dp] = b5[2] * b5[3];
}

// ---------------------------------------------------------------------------
// K3: suppression bitmask. Block = 256 threads (8 wave32), 64 rows per block,
// grid (8, NSETS). Points/areas staged into LDS via TDM tensor_load_to_lds.
// Lane j of a wave computes pair (i, 32w+j); __ballot packs one mask word.
// ---------------------------------------------------------------------------
__global__ void iou_mask_kernel(const float* __restrict__ pts_s,
                                const float* __restrict__ area_s,
                                unsigned* __restrict__ supp) {
  int set = blockIdx.y;
  int rowBase = blockIdx.x * 64;
  __shared__ float s_pts[PP * 8];   // 16 KB
  __shared__ float s_area[PP];      // 2 KB

#if HAVE_TDM
  if (threadIdx.x < 32) {           // one wave issues the DMA
    tdm_load_1d(s_pts, pts_s + (size_t)set * PP * 8, PP * 8);
    tdm_load_1d(s_area, area_s + (size_t)set * PP, PP);
    wait_tensor0();
  }
#else
  for (int e = threadIdx.x; e < PP * 8; e += blockDim.x)
    s_pts[e] = pts_s[(size_t)set * PP * 8 + e];
  for (int e = threadIdx.x; e < PP; e += blockDim.x)
    s_area[e] = area_s[(size_t)set * PP + e];
#endif
  __syncthreads();

  int wave = threadIdx.x >> 5;
  int lane = threadIdx.x & 31;

  for (int r = 0; r < 8; ++r) {
    int i = rowBase + wave * 8 + r;
    float pi[8];
#pragma unroll
    for (int e = 0; e < 8; ++e) pi[e] = s_pts[i * 8 + e];
    float ai = s_area[i];
    for (int w = 0; w < 16; ++w) {
      int j = w * 32 + lane;
      float pj[8];
#pragma unroll
      for (int e = 0; e < 8; ++e) pj[e] = s_pts[j * 8 + e];
      float inter = quad_inter_area(pi, pj);
      float iou = inter / (ai + s_area[j] - inter + 1e-8f);
      unsigned bits = (unsigned)__ballot(iou > NMS_THR);
      if (lane == 0) supp[((size_t)set * PP + i) * 16 + w] = bits;
    }
  }
}

// ---------------------------------------------------------------------------
// K4: greedy NMS bitmask scan — one wave32 per set. Lane k<16 owns keep word k.
// Afterwards emit masked scores msk = keep ? score : -1 (per-set vector).
// ---------------------------------------------------------------------------
__global__ void nms_scan_kernel(const unsigned* __restrict__ supp,
                                const float* __restrict__ sc_s,
                                unsigned* __restrict__ keepw,
                                float* __restrict__ msk) {
  int set = blockIdx.x;
  int lane = threadIdx.x;  // 32
  unsigned kw = 0;
  for (int i = 0; i < PP; ++i) {
    if (i + 8 < PP)
      __builtin_prefetch(&supp[((size_t)set * PP + i + 8) * 16], 0, 0);
    float sv = sc_s[(size_t)set * PP + i];
    unsigned r = (lane < 16) ? supp[((size_t)set * PP + i) * 16 + lane] : 0u;
    bool hit = (r & kw) != 0u;
    bool any = (__ballot(hit) != 0ull);
    bool keep = (sv > 0.0f) && !any;
    if (keep && lane == (i >> 5)) kw |= 1u << (i & 31);
  }
  if (lane < 16) keepw[set * 16 + lane] = kw;

  // broadcast keep words across the wave, emit masked scores
  for (int p = lane; p < PP; p += 32) {
    unsigned w = (unsigned)__shfl((int)kw, p >> 5);
    bool kb = (w >> (p & 31)) & 1u;
    float sv = sc_s[(size_t)set * PP + p];
    msk[(size_t)set * PP + p] = kb ? sv : -1.0f;
  }
}

// ---------------------------------------------------------------------------
// K5: per-image top-100 (stable tie-break by flat index) + output packing.
// The image's masked-score vector (20*512 floats = 40KB) is staged into LDS
// via TDM; the 100 selection sweeps then run entirely out of LDS.
// Output layout (floats): bb[2,100,4] @0 | rr[2,100,5] @800 | sc @1800 | lab @2000
// ---------------------------------------------------------------------------
__global__ void topk_out_kernel(const float* __restrict__ msk,
                                const float* __restrict__ box5_s,
                                const float* __restrict__ bb4_s,
                                float* __restrict__ out) {
  const int img = blockIdx.x;
  const int NFLAT = CFG * PP;  // 10240 per image
  __shared__ float s_msk[CFG * PP];           // 40 KB
  __shared__ unsigned taken[CFG * PP / 32];   // 1.25 KB
  __shared__ float rv[PP];
  __shared__ int   ri[PP];
  __shared__ float selv[DETN];
  __shared__ int   seli[DETN];
  int tid = threadIdx.x;  // 512

#if HAVE_TDM
  if (tid < 32) {
    tdm_load_1d(s_msk, msk + (size_t)img * NFLAT, NFLAT);
    wait_tensor0();
  }
#else
  for (int e = tid; e < NFLAT; e += PP) s_msk[e] = msk[(size_t)img * NFLAT + e];
#endif
  for (int i = tid; i < NFLAT / 32; i += PP) taken[i] = 0u;
  __syncthreads();

  for (int k = 0; k < DETN; ++k) {
    float bv = -1e30f;
    int bi = NFLAT;
    for (int e = tid; e < NFLAT; e += PP) {
      if ((taken[e >> 5] >> (e & 31)) & 1u) continue;
      float v = s_msk[e];
      if (v > bv || (v == bv && e < bi)) { bv = v; bi = e; }
    }
    rv[tid] = bv; ri[tid] = bi;
    __syncthreads();
    for (int s = PP / 2; s > 0; s >>= 1) {
      if (tid < s) {
        float v2 = rv[tid + s]; int i2 = ri[tid + s];
        if (v2 > rv[tid] || (v2 == rv[tid] && i2 < ri[tid])) { rv[tid] = v2; ri[tid] = i2; }
      }
      __syncthreads();
    }
    if (tid == 0) {
      selv[k] = rv[0]; seli[k] = ri[0];
      taken[ri[0] >> 5] |= 1u << (ri[0] & 31);
    }
    __syncthreads();
  }

  for (int k = tid; k < DETN; k += PP) {
    float v = selv[k];
    int e = seli[k];
    bool ok = v > 0.0f;
    int cf = e / PP, p = e - cf * PP;
    int set = img * CFG + cf;
    size_t sp = (size_t)set * PP + p;
#pragma unroll
    for (int j = 0; j < 4; ++j)
      out[(size_t)img * DETN * 4 + k * 4 + j] = ok ? bb4_s[sp * 4 + j] : 0.0f;
#pragma unroll
    for (int j = 0; j < 5; ++j)
      out[BB * DETN * 4 + (size_t)img * DETN * 5 + k * 5 + j] = ok ? box5_s[sp * 5 + j] : 0.0f;
    out[BB * DETN * 9 + (size_t)img * DETN + k] = ok ? v : 0.0f;
    out[BB * DETN * 10 + (size_t)img * DETN + k] = ok ? (float)(cf + 1) : 0.0f;
  }
}

// ---------------------------------------------------------------------------
extern "C" void kernel_launch(void* const* d_in, const int* in_sizes, int n_in,
                              void* d_out, int out_size, void* d_ws, size_t ws_size,
                              hipStream_t stream) {
  (void)in_sizes; (void)n_in; (void)out_size; (void)ws_size;
  const float* logits = (const float*)d_in[0];
  const float* boxreg = (const float*)d_in[1];
  const float* rrects = (const float*)d_in[2];
  float* out = (float*)d_out;

  float* ws = (float*)d_ws;
  float* scores_u = ws;                                  // 40*512
  float* box5_u   = scores_u + (size_t)NSETS * PP;       // 40*512*5
  float* bb4_u    = box5_u + (size_t)NSETS * PP * 5;     // 40*512*4
  float* sc_s     = bb4_u + (size_t)NSETS * PP * 4;      // 40*512
  float* box5_s   = sc_s + (size_t)NSETS * PP;           // 40*512*5
  float* bb4_s    = box5_s + (size_t)NSETS * PP * 5;     // 40*512*4
  float* pts_s    = bb4_s + (size_t)NSETS * PP * 4;      // 40*512*8
  float* area_s   = pts_s + (size_t)NSETS * PP * 8;      // 40*512
  unsigned* supp  = (unsigned*)(area_s + (size_t)NSETS * PP);  // 40*512*16
  unsigned* keepw = supp + (size_t)NSETS * PP * 16;            // 40*16
  float* msk      = (float*)(keepw + NSETS * 16);              // 40*512

  hipLaunchKernelGGL(decode_score_kernel, dim3(BB), dim3(PP), 0, stream,
                     logits, boxreg, rrects, scores_u, box5_u, bb4_u);
  hipLaunchKernelGGL(sort512_kernel, dim3(NSETS), dim3(PP), 0, stream,
                     scores_u, box5_u, bb4_u, sc_s, box5_s, bb4_s, pts_s, area_s);
  hipLaunchKernelGGL(iou_mask_kernel, dim3(8, NSETS), dim3(256), 0, stream,
                     pts_s, area_s, supp);
  hipLaunchKernelGGL(nms_scan_kernel, dim3(NSETS), dim3(32), 0, stream,
                     supp, sc_s, keepw, msk);
  hipLaunchKernelGGL(topk_out_kernel, dim3(BB), dim3(PP), 0, stream,
                     msk, box5_s, bb4_s, out);
}